// UncertaintyDynamicQAgent_11639361372466
// MI455X (gfx1250) — compile-verified
//
#include <hip/hip_runtime.h>
#include <cstdint>
#include <cstddef>

// ---------------------------------------------------------------------------
// UncertaintyDynamicQAgent forward on MI455X (gfx1250).
// One thread == one session; one wave (32 threads) == one block, 128 blocks
// spread the 4096 serial recurrences across the WGPs (latency-bound regime).
// Input chunks are double-buffered in LDS via CDNA5 async global->LDS copies
// (prefetch chunk k+1 while computing chunk k; s_wait_asynccnt 15 retires
// chunk k's in-order loads). Q outputs staged in LDS, written back with
// coalesced 16B/lane b128 stores.
// ---------------------------------------------------------------------------

#define N_TRIALS   1000
#define SPB        32             // sessions (threads) per block == one wave32
#define TC         20             // trials per chunk
#define NCHUNK     (N_TRIALS / TC)
#define INROWF     (TC * 3)       // 60 floats / session / chunk (240 B rows, 16B aligned)
#define OUTROWF    44             // padded out row (40 valid floats + 4 pad => 176 B)
#define IN_PARTS   ((TC * 3 * 4) / 16)   // 15 x 16B per session row
#define OUT_PARTS  ((TC * 2 * 4) / 16)   // 10 x 16B per session row
#define IN_SESS_B  (N_TRIALS * 3 * 4)    // 12000 B global stride per session
#define OUT_SESS_B (N_TRIALS * 2 * 4)    // 8000 B global stride per session

static_assert(SPB == 32, "single-wave blocks: no __syncthreads in async path");

#if defined(__gfx1250__) && __has_builtin(__builtin_amdgcn_global_load_async_to_lds_b128)
#define USE_ASYNC_LDS 1
#else
#define USE_ASYNC_LDS 0
#endif

typedef int v4i __attribute__((ext_vector_type(4)));
typedef __attribute__((address_space(1))) v4i as1_v4i;   // global int4
typedef __attribute__((address_space(3))) v4i as3_v4i;   // LDS int4

#if defined(__gfx1250__) && __has_builtin(__builtin_amdgcn_s_wait_asynccnt)
#define ASYNC_WAIT(n) do { __builtin_amdgcn_s_wait_asynccnt(n); \
                           __atomic_signal_fence(__ATOMIC_SEQ_CST); } while (0)
#else
#define ASYNC_WAIT(n) do { asm volatile("s_wait_asynccnt %0" :: "i"(n) : "memory"); } while (0)
#endif

__device__ __forceinline__ float softplus_f(float z) {
  // Numerically-stable softplus, matches jax.nn.softplus: max(z,0)+log1p(exp(-|z|))
  return fmaxf(z, 0.0f) + log1pf(expf(-fabsf(z)));
}

__device__ __forceinline__ float smooth_clamp01(float x) {
  const float beta = 100.0f;                 // (hi - lo) * 100 with lo=0, hi=1
  float lo_b = softplus_f(x * beta) / beta;  // lo + softplus((x-lo)*beta)/beta
  float hi_b = 1.0f - softplus_f((1.0f - x) * beta) / beta;
  return (x < 0.5f) ? lo_b : hi_b;
}

__device__ __forceinline__ float sigmoid_f(float x) {
  return 1.0f / (1.0f + expf(-x));
}

#if USE_ASYNC_LDS
// Issue 15 coalesced 16B async copies per lane: 20 trials x 32 sessions.
__device__ __forceinline__ void stage_chunk_async(const char* gin, int ch,
                                                  float* lbuf, int tid) {
  const char* src0 = gin + (size_t)ch * (TC * 3 * 4);
#pragma unroll
  for (int i = 0; i < IN_PARTS; ++i) {
    int c = i * SPB + tid;          // flat 16B-part index, lanes contiguous
    int r = c / IN_PARTS;           // session row within block
    int p = c - r * IN_PARTS;       // 16B part within row
    const char* src = src0 + (size_t)r * IN_SESS_B + p * 16;
    char*       dst = (char*)lbuf + r * (INROWF * 4) + p * 16;
    __builtin_amdgcn_global_load_async_to_lds_b128(
        (as1_v4i*)(const void*)src, (as3_v4i*)(void*)dst, 0, 0);
  }
}
#endif

__global__ void __launch_bounds__(SPB)
uqa_forward_kernel(const float* __restrict__ inp,
                   const float* __restrict__ p_a_sr, const float* __restrict__ p_a_su,
                   const float* __restrict__ p_g_lr, const float* __restrict__ p_g_lu,
                   const float* __restrict__ p_g_ar, const float* __restrict__ p_g_au,
                   const float* __restrict__ p_k_sr, const float* __restrict__ p_k_su,
                   const float* __restrict__ p_k_dr, const float* __restrict__ p_k_du,
                   float* __restrict__ out) {
  __shared__ float lin[2][SPB * INROWF];  // 2 x 7680 B double-buffered input
  __shared__ float lout[SPB * OUTROWF];   // 5632 B Q staging

  const int tid = threadIdx.x;
  const int s0  = blockIdx.x * SPB;

  // --- broadcast scalar parameters (uniform -> SALU float pipe) ---
  const float ar  = sigmoid_f(p_a_sr[0]);   // alpha0[0] (== alpha0[2])
  const float au  = sigmoid_f(p_a_su[0]);   // alpha0[1] (== alpha0[3])
  const float gar = sigmoid_f(p_g_ar[0]);   // gamma_a[0]
  const float gau = sigmoid_f(p_g_au[0]);   // gamma_a[1]
  const float glr = sigmoid_f(p_g_lr[0]);   // gamma_l[0] (== gamma_l[2])
  const float glu = sigmoid_f(p_g_lu[0]);   // gamma_l[1] (== gamma_l[3])
  const float k0  = p_k_sr[0];
  const float k1  = p_k_su[0];
  const float k2  = p_k_dr[0];
  const float k3  = p_k_du[0];

  // --- per-session recurrent state (registers) ---
  float Q0 = 0.0f, Q1 = 0.0f;
  float l0 = 0.5f, l1 = 0.5f;
  float alpha = 0.0f;

  const char* gin  = (const char*)inp + (size_t)s0 * IN_SESS_B;
  char*       gout = (char*)out       + (size_t)s0 * OUT_SESS_B;

#if USE_ASYNC_LDS
  stage_chunk_async(gin, 0, lin[0], tid);   // prefetch first chunk
#endif

#pragma unroll 1
  for (int ch = 0; ch < NCHUNK; ++ch) {
    float* cur = lin[ch & 1];

#if USE_ASYNC_LDS
    if (ch + 1 < NCHUNK) {
      stage_chunk_async(gin, ch + 1, lin[(ch + 1) & 1], tid);  // prefetch next
      ASYNC_WAIT(IN_PARTS);   // in-order completion: <=15 outstanding => chunk ch done
    } else {
      ASYNC_WAIT(0);          // last chunk: drain everything
    }
#else
    {
      const char* src0 = gin + (size_t)ch * (TC * 3 * 4);
#pragma unroll
      for (int i = 0; i < IN_PARTS; ++i) {
        int c = i * SPB + tid;
        int r = c / IN_PARTS;
        int p = c - r * IN_PARTS;
        *(float4*)((char*)cur + r * (INROWF * 4) + p * 16) =
            *(const float4*)(src0 + (size_t)r * IN_SESS_B + p * 16);
      }
      __syncthreads();
    }
#endif

    // ---- 20 sequential trial updates from LDS (serial dependence in j) ----
#pragma unroll 4
    for (int j = 0; j < TC; ++j) {
      const float cl = cur[tid * INROWF + j * 3 + 0];
      const float cr = cur[tid * INROWF + j * 3 + 1];
      const float o  = cur[tid * INROWF + j * 3 + 2];
      const float no = 1.0f - o;

      // sel_l = [cl*o, cl*(1-o), cr*o, cr*(1-o)]; sel_r = sel_l[perm(2,3,0,1)]
      const float sl0 = cl * o,  sl1 = cl * no;
      const float sl2 = cr * o,  sl3 = cr * no;

      const float ld0 = k0 - Q0, ld1 = k1 - Q0, ld2 = k2 - Q0, ld3 = k3 - Q0;
      const float rd0 = k0 - Q1, rd1 = k1 - Q1, rd2 = k2 - Q1, rd3 = k3 - Q1;

      const float ll0 = fabsf(ld0) - l0, ll1 = fabsf(ld1) - l0;
      const float ll2 = fabsf(ld2) - l0, ll3 = fabsf(ld3) - l0;
      const float rl0 = fabsf(rd0) - l1, rl1 = fabsf(rd1) - l1;
      const float rl2 = fabsf(rd2) - l1, rl3 = fabsf(rd3) - l1;

      const float a_init = ar * sl0 + au * sl1 + ar * sl2 + au * sl3;
      const float dc = (ar + ll0 - alpha) * sl0 + (au + ll1 - alpha) * sl1
                     + (ar + rl0 - alpha) * sl2 + (au + rl1 - alpha) * sl3;
      const float ge = gar * (sl0 + sl2) + gau * (sl1 + sl3);

      // three independent smooth_clamps -> ILP hides transcendental latency
      const float a_upd = smooth_clamp01(alpha + ge * dc);

      const float ldel = ll0 * glr * sl0 + ll1 * glu * sl1
                       + ll2 * glr * sl2 + ll3 * glu * sl3;
      const float rdel = rl0 * glr * sl2 + rl1 * glu * sl3
                       + rl2 * glr * sl0 + rl3 * glu * sl1;
      l0 = smooth_clamp01(l0 + ldel);
      l1 = smooth_clamp01(l1 + rdel);

      alpha = (ch == 0 && j == 0) ? a_init : a_upd;

      const float qd0 = ld0 * sl0 + ld1 * sl1 + ld2 * sl2 + ld3 * sl3;
      const float qd1 = rd0 * sl2 + rd1 * sl3 + rd2 * sl0 + rd3 * sl1;
      Q0 += alpha * qd0;
      Q1 += alpha * qd1;

      lout[tid * OUTROWF + j * 2 + 0] = Q0;
      lout[tid * OUTROWF + j * 2 + 1] = Q1;
    }

#if !USE_ASYNC_LDS
    __syncthreads();
#endif

    // ---- coalesced write-back of Q trajectory (16B/lane) ----
    {
      char* dst0 = gout + (size_t)ch * (TC * 2 * 4);
#pragma unroll
      for (int i = 0; i < OUT_PARTS; ++i) {
        int c = i * SPB + tid;
        int r = c / OUT_PARTS;
        int p = c - r * OUT_PARTS;
        const float4 v = *(const float4*)((const char*)lout + r * (OUTROWF * 4) + p * 16);
        *(float4*)(dst0 + (size_t)r * OUT_SESS_B + p * 16) = v;
      }
    }
    // single-wave block: in-wave LDS ordering (ds reads above before next
    // chunk's ds writes) is guaranteed by hardware in-order LDS execution
  }
}

extern "C" void kernel_launch(void* const* d_in, const int* in_sizes, int n_in,
                              void* d_out, int out_size, void* d_ws, size_t ws_size,
                              hipStream_t stream) {
  const float* inp  = (const float*)d_in[0];
  const float* a_sr = (const float*)d_in[1];
  const float* a_su = (const float*)d_in[2];
  const float* g_lr = (const float*)d_in[3];
  const float* g_lu = (const float*)d_in[4];
  const float* g_ar = (const float*)d_in[5];
  const float* g_au = (const float*)d_in[6];
  const float* k_sr = (const float*)d_in[7];
  const float* k_su = (const float*)d_in[8];
  const float* k_dr = (const float*)d_in[9];
  const float* k_du = (const float*)d_in[10];
  float* out = (float*)d_out;

  const int n_sess  = in_sizes[0] / (N_TRIALS * 3);  // 4096
  const int nblocks = n_sess / SPB;                  // 128 single-wave blocks

  uqa_forward_kernel<<<nblocks, SPB, 0, stream>>>(
      inp, a_sr, a_su, g_lr, g_lu, g_ar, g_au, k_sr, k_su, k_dr, k_du, out);
}